// BiPartiteAssignmentCriterion_29781303231231
// MI455X (gfx1250) — compile-verified
//
#include <hip/hip_runtime.h>
#include <stdint.h>
#include <stddef.h>

#define BN   256
#define TN   21
#define VN   32000
#define TOK_EOS 2
#define NROW (BN * TN)   /* 5376 = 64 * 84 */

// ---------------- CDNA5 feature detection (device pass only) ----------------
#if defined(__HIP_DEVICE_COMPILE__) && \
    __has_builtin(__builtin_amdgcn_global_load_async_to_lds_b128) && \
    __has_builtin(__builtin_amdgcn_s_wait_asynccnt)
#define HAVE_ASYNC 1
#else
#define HAVE_ASYNC 0
#endif

#if defined(__HIP_DEVICE_COMPILE__) && \
    __has_builtin(__builtin_amdgcn_wmma_f32_16x16x4_f32)
#define HAVE_WMMA4 1
#else
#define HAVE_WMMA4 0
#endif

#if HAVE_ASYNC
typedef int v4i_ __attribute__((vector_size(16)));
typedef __attribute__((address_space(1))) v4i_ gv4i;   // global int4
typedef __attribute__((address_space(3))) v4i_ lv4i;   // LDS int4
__device__ __forceinline__ void async_cp16(const float* g, float* l) {
  // GLOBAL_LOAD_ASYNC_TO_LDS_B128: 16 bytes per lane, global -> LDS, ASYNCcnt
  __builtin_amdgcn_global_load_async_to_lds_b128(
      (gv4i*)(const_cast<float*>(g)), (lv4i*)l, 0, 0);
}
#endif

// =====================================================================
// Kernel 0: per-sample Hungarian matching (one wave32 per sample).
// float64 math, identical op order to the host reference -> identical result.
// Inner column scan is lane-parallel; (min,argmin) via xor-shuffle butterfly
// with tie -> smallest column index (matches sequential "<" scan).
// =====================================================================
__global__ void __launch_bounds__(32) k_match(
    const float* __restrict__ logits, const int* __restrict__ tgt,
    int* __restrict__ matched, int* __restrict__ eosidx)
{
  __shared__ double cmat[TN][TN];
  __shared__ double u[TN + 1], v[TN + 1], minv[TN + 1];
  __shared__ int p[TN + 1], way[TN + 1], used[TN + 1];
  __shared__ int tgs[TN], perm[TN];
  __shared__ int Ls;

  const int i = blockIdx.x;
  const int lane = threadIdx.x;

  if (lane < TN) { tgs[lane] = tgt[i * TN + lane]; perm[lane] = 0; }
  __syncthreads();
  if (lane == 0) {
    int e = 0;
    for (int t = 0; t < TN; ++t) { if (tgs[t] == TOK_EOS) { e = t; break; } }
    Ls = e;
  }
  __syncthreads();
  const int L = Ls;

  const float* rowp = logits + (size_t)i * (size_t)(TN * VN);
  for (int idx = lane; idx < L * L; idx += 32) {
    const int r = idx / L, cc = idx - r * L;
    cmat[r][cc] = -(double)rowp[(size_t)r * VN + tgs[cc]];
  }
  if (lane <= TN) { u[lane] = 0.0; v[lane] = 0.0; p[lane] = 0; way[lane] = 0; }
  __syncthreads();

  const double INFD = 1e18;
  for (int ii = 1; ii <= L; ++ii) {
    if (lane == 0) p[0] = ii;
    if (lane <= L) { minv[lane] = INFD; used[lane] = 0; }
    __syncthreads();
    int j0 = 0;
    for (int guard = 0; guard < 4 * TN; ++guard) {
      if (lane == 0) used[j0] = 1;
      __syncthreads();
      const int i0 = p[j0];
      const double ui0 = u[i0];
      double myv = INFD; int myj = 0x7fffffff;
      if (lane >= 1 && lane <= L && !used[lane]) {
        const double cur = cmat[i0 - 1][lane - 1] - ui0 - v[lane];
        if (cur < minv[lane]) { minv[lane] = cur; way[lane] = j0; }
        myv = minv[lane]; myj = lane;
      }
      for (int off = 16; off; off >>= 1) {
        const double ov = __shfl_xor(myv, off, 32);
        const int oj = __shfl_xor(myj, off, 32);
        if (ov < myv || (ov == myv && oj < myj)) { myv = ov; myj = oj; }
      }
      const double delta = myv; const int j1 = myj;
      __syncthreads();
      if (lane <= L) {
        if (used[lane]) { u[p[lane]] += delta; v[lane] -= delta; }
        else            { minv[lane] -= delta; }
      }
      __syncthreads();
      if (j1 > L) break;   // safety (cannot trigger on well-formed input)
      j0 = j1;
      if (p[j0] == 0) break;
    }
    if (lane == 0) {       // augmenting-path flip (serial, tiny)
      int jj = j0;
      while (jj) { const int jp = way[jj]; p[jj] = p[jp]; jj = jp; }
    }
    __syncthreads();
  }

  if (lane >= 1 && lane <= L && p[lane] > 0) perm[p[lane] - 1] = lane - 1;
  __syncthreads();
  if (lane < TN) {
    int mt = 0;
    if (lane < L) mt = tgs[perm[lane]];
    matched[i * TN + lane] = mt;
  }
  if (lane == 0) eosidx[i] = L;
}

// =====================================================================
// Kernel 1: per-row online log-softmax + NLL. HBM-bound (688 MB stream).
// Logits staged via double-buffered async global->LDS copies (ASYNCcnt),
// each lane consumes only its own 16B so only per-wave s_wait_asynccnt
// is needed. Deterministic block reduction.
// =====================================================================
#define ONLINE_UPD(x) do {                                              \
    const float mx_ = fmaxf(fmaxf((x).x, (x).y), fmaxf((x).z, (x).w)); \
    if (mx_ > m) { s *= __expf(m - mx_); m = mx_; }                     \
    s += __expf((x).x - m) + __expf((x).y - m) +                        \
         __expf((x).z - m) + __expf((x).w - m);                         \
  } while (0)

__global__ void __launch_bounds__(256) k_rowloss(
    const float* __restrict__ logits, const int* __restrict__ matched,
    const int* __restrict__ eosidx, float* __restrict__ labelval,
    float* __restrict__ eosval)
{
  __shared__ float wmred[8], wsred[8];
  const int row = blockIdx.x;
  const int tid = threadIdx.x;
  const float* rowp = logits + (size_t)row * VN;

  float m = -3.0e38f, s = 0.0f;

#if HAVE_ASYNC
  __shared__ float stage[2][1024] __attribute__((aligned(16)));
  async_cp16(rowp + tid * 4, &stage[0][tid * 4]);           // chunk 0
  for (int k = 0; k < 31; ++k) {
    if (k + 1 < 31) {
      async_cp16(rowp + (k + 1) * 1024 + tid * 4, &stage[(k + 1) & 1][tid * 4]);
      __builtin_amdgcn_s_wait_asynccnt(1);                  // chunk k landed
    } else {
      __builtin_amdgcn_s_wait_asynccnt(0);
    }
    asm volatile("" ::: "memory");                          // compiler fence
    const float4 x = *(const float4*)&stage[k & 1][tid * 4];
    ONLINE_UPD(x);
  }
  if (tid < 64) {                                           // tail: 256 floats
    const float4 x = *(const float4*)(rowp + 31 * 1024 + tid * 4);
    ONLINE_UPD(x);
  }
#else
  for (int k4 = tid; k4 < VN / 4; k4 += 256) {
    const float4 x = ((const float4*)rowp)[k4];
    ONLINE_UPD(x);
  }
#endif

  // wave reduce (m,s)
  for (int off = 16; off; off >>= 1) {
    const float om = __shfl_xor(m, off, 32), os = __shfl_xor(s, off, 32);
    const float M = fmaxf(m, om);
    s = s * __expf(m - M) + os * __expf(om - M);
    m = M;
  }
  if ((tid & 31) == 0) { wmred[tid >> 5] = m; wsred[tid >> 5] = s; }
  __syncthreads();
  if (tid < 32) {
    float mm = (tid < 8) ? wmred[tid] : -3.0e38f;
    float ss = (tid < 8) ? wsred[tid] : 0.0f;
    for (int off = 4; off; off >>= 1) {
      const float om = __shfl_xor(mm, off, 32), os = __shfl_xor(ss, off, 32);
      const float M = fmaxf(mm, om);
      ss = ss * __expf(mm - M) + os * __expf(om - M);
      mm = M;
    }
    if (tid == 0) {
      const int i = row / TN, t = row - i * TN;
      const int L = eosidx[i];
      const int tok = (t < L) ? matched[row] : 0;
      const float nll = (mm + __logf(ss)) - rowp[tok];
      labelval[row] = (t < L) ? nll : 0.0f;
      eosval[row]   = (t < L) ? 0.0f : nll;
    }
  }
}

// =====================================================================
// Kernel 2: deterministic final reduction through the matrix pipe.
// v_wmma_f32_16x16x4_f32 with all-ones B: each D row = sum of 4 A values,
// replicated over 16 columns and accumulated in C -> grand total = sum(D)/16.
// =====================================================================
__global__ void __launch_bounds__(32) k_final(
    const float* __restrict__ labelval, const float* __restrict__ eosval,
    const int* __restrict__ eosidx, float* __restrict__ out)
{
  const int lane = threadIdx.x;
  float sl, se;
#if HAVE_WMMA4
  typedef float v2f __attribute__((ext_vector_type(2)));
  typedef float v8f __attribute__((ext_vector_type(8)));
  v8f accL = {0.f, 0.f, 0.f, 0.f, 0.f, 0.f, 0.f, 0.f};
  v8f accE = {0.f, 0.f, 0.f, 0.f, 0.f, 0.f, 0.f, 0.f};
  v2f ones; ones.x = 1.0f; ones.y = 1.0f;
  for (int base = 0; base < NROW; base += 64) {   // 84 iterations, 64 vals each
    v2f a, b;
    a.x = labelval[base + lane]; a.y = labelval[base + 32 + lane];
    b.x = eosval[base + lane];   b.y = eosval[base + 32 + lane];
    accL = __builtin_amdgcn_wmma_f32_16x16x4_f32(false, a, false, ones,
                                                 (short)0, accL, false, false);
    accE = __builtin_amdgcn_wmma_f32_16x16x4_f32(false, b, false, ones,
                                                 (short)0, accE, false, false);
  }
  sl = 0.f; se = 0.f;
  for (int q = 0; q < 8; ++q) { sl += accL[q]; se += accE[q]; }
  sl *= 0.0625f; se *= 0.0625f;   // each row-sum replicated over 16 columns
#else
  sl = 0.f; se = 0.f;
  for (int idx = lane; idx < NROW; idx += 32) { sl += labelval[idx]; se += eosval[idx]; }
#endif
  for (int off = 16; off; off >>= 1) {
    sl += __shfl_xor(sl, off, 32);
    se += __shfl_xor(se, off, 32);
  }
  int cnt = 0;
  for (int idx = lane; idx < BN; idx += 32) cnt += eosidx[idx];
  for (int off = 16; off; off >>= 1) cnt += __shfl_xor(cnt, off, 32);
  if (lane == 0) {
    out[0] = sl / (float)cnt;            // label_loss
    out[1] = se / (float)(NROW - cnt);   // eos_loss
  }
}

// =====================================================================
extern "C" void kernel_launch(void* const* d_in, const int* in_sizes, int n_in,
                              void* d_out, int out_size, void* d_ws, size_t ws_size,
                              hipStream_t stream)
{
  (void)in_sizes; (void)n_in; (void)out_size; (void)ws_size;
  const float* logits  = (const float*)d_in[0];
  const int*   targets = (const int*)d_in[1];
  float* out = (float*)d_out;

  float* labelval = (float*)d_ws;            // [NROW]
  float* eosval   = labelval + NROW;         // [NROW]
  int*   matched  = (int*)(eosval + NROW);   // [NROW]
  int*   eosidx   = matched + NROW;          // [BN]

  k_match  <<<BN,   32,  0, stream>>>(logits, targets, matched, eosidx);
  k_rowloss<<<NROW, 256, 0, stream>>>(logits, matched, eosidx, labelval, eosval);
  k_final  <<<1,    32,  0, stream>>>(labelval, eosval, eosidx, out);
}